// SSMLayer_31155692765735
// MI455X (gfx1250) — compile-verified
//
#include <hip/hip_runtime.h>

// ---------------- problem constants ----------------
#define D_MODEL 512
#define D_STATE 64
#define BATCH   8
#define SEQ     4096
#define MROWS   (BATCH * SEQ)       // 32768 rows
#define NCHUNK  64                  // scan chunks per sequence
#define CHUNK_L (SEQ / NCHUNK)      // 64 steps per chunk

typedef __attribute__((ext_vector_type(2))) float v2f;
typedef __attribute__((ext_vector_type(4))) float v4f;
typedef __attribute__((ext_vector_type(8))) float v8f;

// ---------------- LayerNorm (one row of 512 per 256-thread block) -------
__global__ __launch_bounds__(256) void ln_kernel(
    const float* __restrict__ x, const float* __restrict__ gamma,
    const float* __restrict__ beta, float* __restrict__ xn) {
  const int row = blockIdx.x;
  const int tid = threadIdx.x;
  const float* xr = x + (size_t)row * D_MODEL;
  const float v0 = xr[tid];
  const float v1 = xr[tid + 256];
  float s  = v0 + v1;
  float s2 = v0 * v0 + v1 * v1;
#pragma unroll
  for (int off = 16; off > 0; off >>= 1) {
    s  += __shfl_xor(s,  off, 32);
    s2 += __shfl_xor(s2, off, 32);
  }
  __shared__ float shs[8], shq[8];
  const int lane = tid & 31, wid = tid >> 5;
  if (lane == 0) { shs[wid] = s; shq[wid] = s2; }
  __syncthreads();
  s = 0.f; s2 = 0.f;
#pragma unroll
  for (int w = 0; w < 8; ++w) { s += shs[w]; s2 += shq[w]; }
  const float mean = s * (1.0f / D_MODEL);
  const float var  = s2 * (1.0f / D_MODEL) - mean * mean;
  const float rstd = rsqrtf(var + 1e-3f);
  float* outr = xn + (size_t)row * D_MODEL;
  outr[tid]       = (v0 - mean) * rstd * gamma[tid]       + beta[tid];
  outr[tid + 256] = (v1 - mean) * rstd * gamma[tid + 256] + beta[tid + 256];
}

// ---------------- fp32 WMMA GEMM: out[M,Nc] = A[M,K] @ W[K,Nc] (+epilogue)
// Block tile 128x64, 8 waves x (16 rows x 64 cols), K staged 32-wide in LDS.
// W is stored K-pair-interleaved in LDS so each B fragment is one ds_load_b64.
// K-panel staging is software-pipelined through registers; all staging
// addresses are affine per-thread bases to avoid 64-bit VALU muls in the loop.
__global__ __launch_bounds__(256) void wmma_gemm_f32(
    const float* __restrict__ A, const float* __restrict__ W,
    const float* __restrict__ bias,
    const float* __restrict__ add, const float* __restrict__ addScale,
    float* __restrict__ out, int K, int Nc) {
  __shared__ float lA[128 * 36];   // 128 rows x 32 cols, stride 36 (bank-safe, 16B aligned)
  __shared__ float lW[16 * 160];   // 16 K-pairs x 64 cols x 2, stride 160 (half-waves disjoint)

  const int tid  = threadIdx.x;
  const int n0   = blockIdx.x * 64;
  const int m0   = blockIdx.y * 128;
  const int lane = tid & 31, wid = tid >> 5;
  const int l15  = lane & 15, lh = lane >> 4;  // lh selects K-pair / M+8 half
  const int wm   = wid * 16;                   // wave's 16-row strip

  v8f acc[4];
#pragma unroll
  for (int t = 0; t < 4; ++t)
#pragma unroll
    for (int r = 0; r < 8; ++r) acc[t][r] = 0.0f;

  // ---- per-thread staging bases (affine; no 64-bit muls in the hot loop) --
  const int lr    = tid >> 1;          // A-tile row this thread loads
  const int lhalf = (tid & 1) * 16;    // which 16-col half
  const float* aRow = A + (size_t)(m0 + lr) * K + lhalf;

  const int kr0 = tid >> 6;            // W K-row base (0..3), +4 per j
  const int nc  = tid & 63;            // W column (fixed per thread)
  const float* wCur = W + (size_t)kr0 * Nc + n0 + nc;
  float* lwBase = &lW[(kr0 >> 1) * 160 + nc * 2 + (kr0 & 1)];  // + j*320
  v4f*   laBase = reinterpret_cast<v4f*>(&lA[lr * 36 + lhalf]);

  auto load_panel = [&](int kt, v4f (&ra)[4], float (&rw)[8]) {
    const v4f* s4 = reinterpret_cast<const v4f*>(aRow + kt);
#pragma unroll
    for (int j = 0; j < 4; ++j) ra[j] = s4[j];
    const float* wp = wCur + (size_t)kt * Nc;
#pragma unroll
    for (int j = 0; j < 8; ++j) rw[j] = wp[(size_t)(4 * j) * Nc];
  };
  auto store_panel = [&](const v4f (&ra)[4], const float (&rw)[8]) {
#pragma unroll
    for (int j = 0; j < 4; ++j) laBase[j] = ra[j];
#pragma unroll
    for (int j = 0; j < 8; ++j) lwBase[j * 320] = rw[j];  // pair-interleaved
  };

  v4f ra[4]; float rw[8];
  load_panel(0, ra, rw);
  store_panel(ra, rw);
  __syncthreads();

  for (int kt = 0; kt < K; kt += 32) {
    const bool notlast = (kt + 32 < K);
    if (notlast) {
      load_panel(kt + 32, ra, rw);                       // overlap with WMMAs below
      if (kt + 64 < K) __builtin_prefetch(aRow + kt + 64, 0, 0);
    }

#pragma unroll
    for (int k = 0; k < 32; k += 4) {
      // A frag 16x4: lanes0-15 -> K{k,k+1}; lanes16-31 -> K{k+2,k+3}
      const v2f af = *reinterpret_cast<const v2f*>(
          &lA[(wm + l15) * 36 + k + lh * 2]);
#pragma unroll
      for (int t = 0; t < 4; ++t) {
        // B frag: rows {k+2lh, k+2lh+1} at col -> adjacent floats in lW
        const v2f bf = *reinterpret_cast<const v2f*>(
            &lW[((k >> 1) + lh) * 160 + (t * 16 + l15) * 2]);
        acc[t] = __builtin_amdgcn_wmma_f32_16x16x4_f32(
            false, af, false, bf, (short)0, acc[t], false, false);
      }
    }

    __syncthreads();                 // all waves done reading this panel
    if (notlast) {
      store_panel(ra, rw);
      __syncthreads();               // panel visible to all waves
    }
  }

  // epilogue: + bias[n] + addScale[n] * add[m,n]  (pointers may be null)
  const size_t rowBase = (size_t)(m0 + wm + lh * 8) * Nc;
#pragma unroll
  for (int t = 0; t < 4; ++t) {
    const int col = n0 + t * 16 + l15;
    const float b  = bias ? bias[col] : 0.0f;
    const float sc = addScale ? addScale[col] : 1.0f;
#pragma unroll
    for (int r = 0; r < 8; ++r) {
      const size_t off = rowBase + (size_t)r * Nc + col;
      float v = acc[t][r] + b;
      if (add) v += sc * add[off];
      out[off] = v;
    }
  }
}

// ---------------- Bx = Bmat * xs (elementwise) ----------------
__global__ __launch_bounds__(256) void ewmul_kernel(
    const float* __restrict__ a, const float* __restrict__ b,
    float* __restrict__ o, int n) {
  const int i = blockIdx.x * 256 + threadIdx.x;
  if (i < n) o[i] = a[i] * b[i];
}

// ---- scan phase 1: per-chunk local scan (zero init), emit chunk carry ----
__global__ __launch_bounds__(64) void scan1_kernel(
    const float* __restrict__ Bx, const float* __restrict__ A_log,
    float* __restrict__ hs, float* __restrict__ carry) {
  const int n = threadIdx.x;
  const int c = blockIdx.x;
  const int b = blockIdx.y;
  const float a = __expf(-__expf(A_log[n]));
  const size_t base = (((size_t)b * SEQ) + (size_t)c * CHUNK_L) * D_STATE + n;
  float h = 0.0f;
  for (int i = 0; i < CHUNK_L; ++i) {
    h = a * h + Bx[base + (size_t)i * D_STATE];
    hs[base + (size_t)i * D_STATE] = h;
  }
  carry[((b * NCHUNK) + c) * D_STATE + n] = h;
}

// ---- scan phase 2: sequential combine of 64 chunk carries per (b,n) ----
__global__ __launch_bounds__(64) void scan2_kernel(
    const float* __restrict__ carry, const float* __restrict__ A_log,
    float* __restrict__ carryIn) {
  const int n = threadIdx.x;
  const int b = blockIdx.x;
  const float a = __expf(-__expf(A_log[n]));
  float aL = a;
#pragma unroll
  for (int j = 0; j < 6; ++j) aL *= aL;   // a^64 exactly by squaring
  float run = 0.0f;
  for (int c = 0; c < NCHUNK; ++c) {
    const size_t idx = ((size_t)(b * NCHUNK) + c) * D_STATE + n;
    carryIn[idx] = run;                   // exclusive carry into chunk c
    run = aL * run + carry[idx];
  }
}

// ---- scan phase 3 fused with G = Cmat * h: h = hs_local + carry * a^(i+1)
__global__ __launch_bounds__(256) void gfuse_kernel(
    const float* __restrict__ hs, const float* __restrict__ carryIn,
    const float* __restrict__ Cmat, const float* __restrict__ A_log,
    float* __restrict__ G) {
  const int tid = threadIdx.x;
  const int c = blockIdx.x;
  const int b = blockIdx.y;
  const int n  = tid & 63;
  const int i0 = tid >> 6;                // 0..3
  const float a  = __expf(-__expf(A_log[n]));
  const float ci = carryIn[((b * NCHUNK) + c) * D_STATE + n];
  float p = a;
  for (int j = 0; j < i0; ++j) p *= a;    // a^(i0+1)
  const float a2 = a * a;
  const float a4 = a2 * a2;
  const size_t base = (((size_t)b * SEQ) + (size_t)c * CHUNK_L) * D_STATE + n;
  for (int i = i0; i < CHUNK_L; i += 4) {
    const size_t idx = base + (size_t)i * D_STATE;
    const float h = hs[idx] + ci * p;
    G[idx] = Cmat[idx] * h;
    p *= a4;
  }
}

// ---------------- launch ----------------
extern "C" void kernel_launch(void* const* d_in, const int* in_sizes, int n_in,
                              void* d_out, int out_size, void* d_ws, size_t ws_size,
                              hipStream_t stream) {
  (void)in_sizes; (void)n_in; (void)out_size; (void)ws_size;
  const float* x      = (const float*)d_in[0];
  const float* gamma  = (const float*)d_in[1];
  const float* beta   = (const float*)d_in[2];
  const float* W_in   = (const float*)d_in[3];
  const float* b_in   = (const float*)d_in[4];
  const float* W_xs   = (const float*)d_in[5];
  const float* W_B    = (const float*)d_in[6];
  const float* b_B    = (const float*)d_in[7];
  const float* W_C    = (const float*)d_in[8];
  const float* b_C    = (const float*)d_in[9];
  const float* A_log  = (const float*)d_in[10];
  const float* D_skip = (const float*)d_in[11];
  const float* W_so   = (const float*)d_in[12];
  const float* W_out  = (const float*)d_in[13];
  const float* b_out  = (const float*)d_in[14];
  float* out = (float*)d_out;

  float* ws = (float*)d_ws;
  const size_t MD = (size_t)MROWS * D_MODEL;   // 16.78M floats
  const size_t MN = (size_t)MROWS * D_STATE;   //  2.10M floats
  float* xn   = ws;             // [M,512] xn, later reused as y
  float* xp   = ws + MD;        // [M,512]
  float* bufA = ws + 2 * MD;    // [M,64]  Bmat -> Bx -> G
  float* bufC = bufA + MN;      // [M,64]  Cmat
  float* bufS = bufA + 2 * MN;  // [M,64]  xs -> hs
  float* carry   = bufA + 3 * MN;                      // [B,64,64]
  float* carryIn = carry + (size_t)BATCH * NCHUNK * D_STATE;

  const dim3 blk(256);

  // 1) LayerNorm
  ln_kernel<<<MROWS, blk, 0, stream>>>(x, gamma, beta, xn);
  // 2) xp = xn @ W_in + b_in
  wmma_gemm_f32<<<dim3(D_MODEL / 64, MROWS / 128), blk, 0, stream>>>(
      xn, W_in, b_in, nullptr, nullptr, xp, D_MODEL, D_MODEL);
  // 3) Bmat = xp @ W_B + b_B
  wmma_gemm_f32<<<dim3(1, MROWS / 128), blk, 0, stream>>>(
      xp, W_B, b_B, nullptr, nullptr, bufA, D_MODEL, D_STATE);
  // 4) Cmat = xp @ W_C + b_C
  wmma_gemm_f32<<<dim3(1, MROWS / 128), blk, 0, stream>>>(
      xp, W_C, b_C, nullptr, nullptr, bufC, D_MODEL, D_STATE);
  // 5) xs = xp @ W_xs
  wmma_gemm_f32<<<dim3(1, MROWS / 128), blk, 0, stream>>>(
      xp, W_xs, nullptr, nullptr, nullptr, bufS, D_MODEL, D_STATE);
  // 6) Bx = Bmat * xs (in place over Bmat)
  ewmul_kernel<<<(MROWS * D_STATE) / 256, blk, 0, stream>>>(
      bufA, bufS, bufA, MROWS * D_STATE);
  // 7) chunked scan: local scans -> chunk carries (hs overwrites xs buffer)
  scan1_kernel<<<dim3(NCHUNK, BATCH), 64, 0, stream>>>(bufA, A_log, bufS, carry);
  // 8) combine chunk carries (tiny sequential pass)
  scan2_kernel<<<BATCH, 64, 0, stream>>>(carry, A_log, carryIn);
  // 9) G = Cmat * (hs + carry * a^(i+1))  (G overwrites Bx buffer)
  gfuse_kernel<<<dim3(NCHUNK, BATCH), blk, 0, stream>>>(
      bufS, carryIn, bufC, A_log, bufA);
  // 10) y = G @ W_so + D_skip * xp   (y reuses xn buffer)
  wmma_gemm_f32<<<dim3(D_MODEL / 64, MROWS / 128), blk, 0, stream>>>(
      bufA, W_so, nullptr, xp, D_skip, xn, D_STATE, D_MODEL);
  // 11) out = y @ W_out + b_out + residual
  wmma_gemm_f32<<<dim3(D_MODEL / 64, MROWS / 128), blk, 0, stream>>>(
      xn, W_out, b_out, x, nullptr, out, D_MODEL, D_MODEL);
}